// Encoder_68564857914108
// MI455X (gfx1250) — compile-verified
//
#include <hip/hip_runtime.h>
#include <hip/hip_bf16.h>
#include <math.h>

// Problem constants (from reference)
static constexpr int BB = 16, SS = 12, NN = 2048, IND = 2;
static constexpr int HH = 64, HEADS = 4, DH = 16;
static constexpr int EE = 32768;
static constexpr float DTc = 0.1f;
static constexpr int STEPS = 12;
static constexpr float NEG = 0.2f;
static constexpr int RR = BB * NN;          // 32768 flattened (b,n) rows
static constexpr int ELEMS = RR * HH;       // 2,097,152

typedef __attribute__((ext_vector_type(2))) float v2f;
typedef __attribute__((ext_vector_type(8))) float v8f;

// ---------------- CSR build over dst (deterministic, no atomics) ----------------
__global__ void k_count(const int* __restrict__ dst, int* __restrict__ counts) {
    int n = blockIdx.x * blockDim.x + threadIdx.x;
    if (n >= NN) return;
    int c = 0;
    for (int e = 0; e < EE; ++e) c += (dst[e] == n) ? 1 : 0;
    counts[n] = c;
}

__global__ void k_scan(const int* __restrict__ counts, int* __restrict__ offs) {
    if (blockIdx.x == 0 && threadIdx.x == 0) {
        int run = 0;
        for (int n = 0; n < NN; ++n) { offs[n] = run; run += counts[n]; }
        offs[NN] = run;
    }
}

__global__ void k_fill(const int* __restrict__ dst, const int* __restrict__ offs,
                       int* __restrict__ eidx) {
    int n = blockIdx.x * blockDim.x + threadIdx.x;
    if (n >= NN) return;
    int idx = offs[n];
    for (int e = 0; e < EE; ++e)
        if (dst[e] == n) eidx[idx++] = e;
}

// ---------------- y0 = inputs[:,0] @ w_in + b_in ----------------
__global__ void k_init_y(const float* __restrict__ inputs, const float* __restrict__ w_in,
                         const float* __restrict__ b_in, float* __restrict__ y) {
    int gid = blockIdx.x * blockDim.x + threadIdx.x;
    if (gid >= ELEMS) return;
    int r = gid >> 6, hc = gid & 63;
    int b = r >> 11, n = r & (NN - 1);
    size_t base = ((size_t)b * SS * NN + n) * IND;   // s = 0 slice
    float x0 = inputs[base + 0], x1 = inputs[base + 1];
    y[gid] = fmaf(x0, w_in[hc], fmaf(x1, w_in[HH + hc], b_in[hc]));
}

// ---------------- WMMA GEMM: out[M x 64] = (A + coeff*Aalt)[M x 64] @ W[64 x 64] ----
// FUSE: read A + coeff*Aalt (RK4 stage input).  MODE 0: plain  1: +bias,tanh  2: +bias
// One wave computes a 32x16 tile: two accumulators share each B fragment, so
// each pair of v_wmma_f32_16x16x4_f32 reuses one B load. 4 waves cover 64 cols.
template <bool FUSE, int MODE>
__global__ __launch_bounds__(128) void k_gemm64(
        const float* __restrict__ A, const float* __restrict__ Aalt, float coeff,
        const float* __restrict__ W, const float* __restrict__ bias,
        float* __restrict__ out) {
    __builtin_prefetch(W, 0, 3);                 // global_prefetch_b8 of 16KB weights
    int lane = threadIdx.x & 31;
    int wave = threadIdx.x >> 5;
    int row0 = blockIdx.x * 32;
    int col0 = wave * 16;                        // 4 waves cover all 64 columns
    int half = lane >> 4;                        // 0: K{0,1} side, 1: K{2,3} side
    int l16  = lane & 15;

    const float* arow0 = A + (size_t)(row0 + l16) * HH;
    const float* arow1 = arow0 + (size_t)16 * HH;
    const float* krow0 = Aalt + (size_t)(row0 + l16) * HH;
    const float* krow1 = krow0 + (size_t)16 * HH;
    const float* wcol  = W + col0 + l16;

    v8f c0 = {}, c1 = {};
#pragma unroll
    for (int kk = 0; kk < 16; ++kk) {
        int kb = kk * 4 + half * 2;
        v2f a0, a1, bf;
        bf.x = wcol[(size_t)kb * HH];            // B: 4x16 f32 fragment (shared)
        bf.y = wcol[(size_t)(kb + 1) * HH];
        a0.x = arow0[kb]; a0.y = arow0[kb + 1];
        a1.x = arow1[kb]; a1.y = arow1[kb + 1];
        if (FUSE) {
            a0.x = fmaf(coeff, krow0[kb],     a0.x);
            a0.y = fmaf(coeff, krow0[kb + 1], a0.y);
            a1.x = fmaf(coeff, krow1[kb],     a1.x);
            a1.y = fmaf(coeff, krow1[kb + 1], a1.y);
        }
        c0 = __builtin_amdgcn_wmma_f32_16x16x4_f32(false, a0, false, bf,
                                                   (short)0, c0, false, false);
        c1 = __builtin_amdgcn_wmma_f32_16x16x4_f32(false, a1, false, bf,
                                                   (short)0, c1, false, false);
    }

    int ncol = col0 + l16;
    float bv = (MODE != 0) ? bias[ncol] : 0.0f;
#pragma unroll
    for (int j = 0; j < 8; ++j) {
        int m = j + half * 8;                    // C/D layout: VGPR j -> M=j / M=j+8
        float v0 = c0[j] + bv;
        float v1 = c1[j] + bv;
        if (MODE == 1) { v0 = tanhf(v0); v1 = tanhf(v1); }
        out[(size_t)(row0 + m) * HH + ncol]      = v0;
        out[(size_t)(row0 + 16 + m) * HH + ncol] = v1;
    }
}

// ---------------- attention logit pieces: e_s/e_d [R x HEADS] ----------------
__global__ void k_es_ed(const float* __restrict__ h, const float* __restrict__ a_src,
                        const float* __restrict__ a_dst,
                        float* __restrict__ es, float* __restrict__ ed) {
    int gid = blockIdx.x * blockDim.x + threadIdx.x;     // R*HEADS
    if (gid >= RR * HEADS) return;
    int r = gid >> 2, head = gid & 3;
    const float* hp = h + (size_t)r * HH + head * DH;
    const float* as = a_src + head * DH;
    const float* ad = a_dst + head * DH;
    float s = 0.f, d = 0.f;
#pragma unroll
    for (int i = 0; i < DH; ++i) { s = fmaf(hp[i], as[i], s); d = fmaf(hp[i], ad[i], d); }
    es[gid] = s; ed[gid] = d;
}

// ---------------- edge softmax + aggregation: one block per (dst node, batch) ----
__global__ __launch_bounds__(64) void k_edge(
        const float* __restrict__ h, const float* __restrict__ es,
        const float* __restrict__ ed, const int* __restrict__ offs,
        const int* __restrict__ eidx, const int* __restrict__ src,
        const float* __restrict__ bias_g, float* __restrict__ kout) {
    int r = blockIdx.x;                  // r = b*N + n
    int b = r >> 11;
    int n = r & (NN - 1);
    int t = threadIdx.x;                 // t = head*16 + dh
    int head = t >> 4;

    int beg = offs[n], end = offs[n + 1];
    float edv = ed[(size_t)r * HEADS + head];

    if (beg == end) {                    // isolated node: agg == 0
        kout[(size_t)r * HH + t] = bias_g[t];
        return;
    }

    float m = -3.402823e38f;
    for (int i = beg; i < end; ++i) {
        int e = eidx[i];
        int s = src[e];
        float x = es[((size_t)b * NN + s) * HEADS + head] + edv;
        float logit = x > 0.f ? x : NEG * x;
        m = fmaxf(m, logit);
    }
    float sumw = 0.f, acc = 0.f;
    for (int i = beg; i < end; ++i) {
        int e = eidx[i];
        int s = src[e];
        float x = es[((size_t)b * NN + s) * HEADS + head] + edv;
        float logit = x > 0.f ? x : NEG * x;
        float w = __expf(logit - m);
        sumw += w;
        acc = fmaf(w, h[((size_t)b * NN + s) * HH + t], acc);
    }
    kout[(size_t)r * HH + t] = acc / sumw + bias_g[t];
}

// ---------------- RK4 bookkeeping ----------------
__global__ void k_acc(float* __restrict__ acc, const float* __restrict__ kv,
                      float w, int set) {
    int gid = blockIdx.x * blockDim.x + threadIdx.x;
    if (gid >= ELEMS) return;
    float v = w * kv[gid];
    acc[gid] = set ? v : acc[gid] + v;
}

__global__ void k_update(float* __restrict__ y, const float* __restrict__ acc) {
    int gid = blockIdx.x * blockDim.x + threadIdx.x;
    if (gid >= ELEMS) return;
    y[gid] += (DTc / 6.0f) * acc[gid];
}

// ---------------- host orchestration ----------------
extern "C" void kernel_launch(void* const* d_in, const int* in_sizes, int n_in,
                              void* d_out, int out_size, void* d_ws, size_t ws_size,
                              hipStream_t stream) {
    const float* inputs = (const float*)d_in[0];
    const float* w_in   = (const float*)d_in[1];
    const float* b_in   = (const float*)d_in[2];
    const float* Wh     = (const float*)d_in[3];
    const float* bias_g = (const float*)d_in[4];
    const float* a_src  = (const float*)d_in[5];
    const float* a_dst  = (const float*)d_in[6];
    const float* w1     = (const float*)d_in[7];
    const float* b1     = (const float*)d_in[8];
    const float* w2     = (const float*)d_in[9];
    const float* b2     = (const float*)d_in[10];
    const int*   src    = (const int*)d_in[11];
    const int*   dst    = (const int*)d_in[12];
    float* outp = (float*)d_out;

    char* ws = (char*)d_ws;
    constexpr size_t SZ = (size_t)ELEMS * sizeof(float);        // 8 MB
    float* y    = (float*)(ws + 0 * SZ);
    float* h    = (float*)(ws + 1 * SZ);    // reused as tanh buffer at the end
    float* kcur = (float*)(ws + 2 * SZ);
    float* acc  = (float*)(ws + 3 * SZ);
    float* es   = (float*)(ws + 4 * SZ);
    float* ed   = (float*)(ws + 4 * SZ + (size_t)RR * HEADS * sizeof(float));
    int* counts = (int*)  (ws + 4 * SZ + 2 * (size_t)RR * HEADS * sizeof(float));
    int* offs   = counts + NN;              // NN+1 entries
    int* eidx   = offs + (NN + 2);          // EE entries

    const int TPB = 256;
    const int NB_ELEMS = (ELEMS + TPB - 1) / TPB;       // 8192
    const int NB_N     = (NN + TPB - 1) / TPB;          // 8
    const int NB_ESED  = (RR * HEADS + TPB - 1) / TPB;  // 512
    const int NB_GEMM  = RR / 32;                       // 1024 row tiles (32 rows each)

    // CSR over dst (fixed for whole integration)
    k_count<<<NB_N, TPB, 0, stream>>>(dst, counts);
    k_scan <<<1, 1, 0, stream>>>(counts, offs);
    k_fill <<<NB_N, TPB, 0, stream>>>(dst, offs, eidx);

    // y0
    k_init_y<<<NB_ELEMS, TPB, 0, stream>>>(inputs, w_in, b_in, y);

    const float stage_c[4] = {0.0f, 0.5f * DTc, 0.5f * DTc, DTc};
    const float stage_w[4] = {1.0f, 2.0f, 2.0f, 1.0f};

    for (int step = 0; step < STEPS; ++step) {
        for (int sIdx = 0; sIdx < 4; ++sIdx) {
            // h = (y + c*k_prev) @ Wh   (k_prev = kcur from previous stage)
            if (sIdx == 0)
                k_gemm64<false, 0><<<NB_GEMM, 128, 0, stream>>>(
                    y, y, 0.0f, Wh, (const float*)nullptr, h);
            else
                k_gemm64<true, 0><<<NB_GEMM, 128, 0, stream>>>(
                    y, kcur, stage_c[sIdx], Wh, (const float*)nullptr, h);
            k_es_ed<<<NB_ESED, TPB, 0, stream>>>(h, a_src, a_dst, es, ed);
            k_edge <<<RR, 64, 0, stream>>>(h, es, ed, offs, eidx, src, bias_g, kcur);
            k_acc  <<<NB_ELEMS, TPB, 0, stream>>>(acc, kcur, stage_w[sIdx],
                                                  sIdx == 0 ? 1 : 0);
        }
        k_update<<<NB_ELEMS, TPB, 0, stream>>>(y, acc);
    }

    // out = tanh(y@w1 + b1) @ w2 + b2
    k_gemm64<false, 1><<<NB_GEMM, 128, 0, stream>>>(y, y, 0.0f, w1, b1, h);
    k_gemm64<false, 2><<<NB_GEMM, 128, 0, stream>>>(h, h, 0.0f, w2, b2, outp);
    (void)in_sizes; (void)n_in; (void)out_size; (void)ws_size;
}